// AttModel_66700842107242
// MI455X (gfx1250) — compile-verified
//
#include <hip/hip_runtime.h>
#include <hip/hip_bf16.h>
#include <math.h>

// ---------------------------------------------------------------------------
// AttModel (motion-prediction attention + DCT GCN) for gfx1250 / MI455X.
// Dominant cost: 4x fused gconv  out = att(66x66) @ (Y(66x512) @ W(512x512))
// via V_WMMA_F32_16X16X4_F32. One workgroup (8 wave32) per batch element,
// Y_b staged in LDS with bank-conflict-free padding, BN+tanh+residual fused.
// K/N are compile-time so W loads use immediate offsets (no 64-bit muls).
// ---------------------------------------------------------------------------

typedef float v2f __attribute__((ext_vector_type(2)));
typedef float v8f __attribute__((ext_vector_type(8)));

#define BN_INV_F 0.9999950000374997f   // float32(1/sqrt(1+1e-5))
#define PI_F 3.14159265358979323846f

static __device__ __forceinline__ v8f wmma_f32(v2f a, v2f b, v8f c) {
  // D = A(16x4,f32) * B(4x16,f32) + C(16x16,f32)
  return __builtin_amdgcn_wmma_f32_16x16x4_f32(false, a, false, b, (short)0, c,
                                               false, false);
}

// ---------------------------------------------------------------------------
// DCT-II matrix (orthonormal => inverse == transpose), 35x35 row-major [k][i]
// ---------------------------------------------------------------------------
__global__ void init_dctm_kernel(float* __restrict__ dctm) {
  int idx = blockIdx.x * blockDim.x + threadIdx.x;
  if (idx < 35 * 35) {
    int k = idx / 35, i = idx % 35;
    float w = (k == 0) ? sqrtf(1.0f / 35.0f) : sqrtf(2.0f / 35.0f);
    dctm[idx] = w * cosf(PI_F * ((float)i + 0.5f) * (float)k / 35.0f);
  }
}

// ---------------------------------------------------------------------------
// Stage 1: attention front-end. One block (256 thr) per batch element.
// ---------------------------------------------------------------------------
__global__ void __launch_bounds__(256) stage1_kernel(
    const float* __restrict__ src,   // (B,50,66)
    const float* __restrict__ qw,    // (1,10)
    const float* __restrict__ qb,    // (1,)
    const float* __restrict__ kw,    // (16,25)
    const float* __restrict__ kb,    // (16,)
    const float* __restrict__ dctm,  // (35,35)
    float* __restrict__ dct_in)      // (B,66,20)
{
  __shared__ float s[50 * 66];
  __shared__ float keyS[66 * 16];
  __shared__ float qS[66];
  __shared__ float attw[16];
  __shared__ float red[20];
  int b = blockIdx.x, tid = threadIdx.x;
  const float* sb = src + (size_t)b * 50 * 66;
  for (int i = tid; i < 50 * 66; i += 256) s[i] = sb[i];
  __syncthreads();

  for (int o = tid; o < 66 * 16; o += 256) {   // key[f,i]
    int f = o >> 4, i = o & 15;
    float a = kb[i];
    for (int t = 0; t < 25; ++t) a += s[t * 66 + f] * kw[i * 25 + t];
    keyS[o] = a;
  }
  for (int f = tid; f < 66; f += 256) {        // query[f]
    float a = qb[0];
    for (int t = 0; t < 10; ++t) a += s[(40 + t) * 66 + f] * qw[t];
    qS[f] = a;
  }
  __syncthreads();
  if (tid < 16) {
    float a = 1e-15f;
    for (int f = 0; f < 66; ++f) a += qS[f] * keyS[f * 16 + tid];
    red[tid] = a;
  }
  __syncthreads();
  if (tid == 0) {
    float ssum = 0.f;
    for (int i = 0; i < 16; ++i) ssum += red[i];
    red[16] = ssum;
  }
  __syncthreads();
  if (tid < 16) attw[tid] = red[tid] / red[16];
  __syncthreads();

  for (int o = tid; o < 660; o += 256) {       // (f,d) outputs
    int f = o / 10, d = o % 10;
    const float* drow = dctm + d * 35;
    float g = 0.f;
    for (int v = 0; v < 35; ++v) {
      int t = (v < 10) ? (40 + v) : 49;
      g += drow[v] * s[t * 66 + f];
    }
    float a = 0.f;
    for (int n = 0; n < 16; ++n) {
      float acc = 0.f;
      for (int v = 0; v < 35; ++v) acc += drow[v] * s[(n + v) * 66 + f];
      a += attw[n] * acc;
    }
    float* dst = dct_in + ((size_t)b * 66 + f) * 20;
    dst[d] = g;
    dst[10 + d] = a;
  }
}

// ---------------------------------------------------------------------------
// Fused gconv:  out[b] = act( att @ (X[b] @ W) + bias )  (+ residual)
// LDS (floats), padded strides for conflict-free b64 access patterns:
//   yP   80 x (K+4)   rows 66..79 zero      (stride%64 == 4 for K=512)
//   attP 80 x 84      zero padded           (84%64 == 20)
//   t    per-wave 80 x 20 slab of X@W       (20: disjoint half-wave windows)
// ---------------------------------------------------------------------------
template <int K, int N, bool BN, bool RES>
__global__ void __launch_bounds__(256) gconv_kernel(
    const float* __restrict__ xin, const float* __restrict__ att,
    const float* __restrict__ W, const float* __restrict__ bias,
    const float* __restrict__ gamma, const float* __restrict__ beta,
    const float* __restrict__ resid, float* __restrict__ out)
{
  constexpr int YS = K + 4;    // yP row stride
  constexpr int AS = 84;       // attP row stride
  constexpr int TS = 20;       // t row stride
  extern __shared__ float lds[];
  float* yP = lds;                   // 80*YS
  float* attP = yP + 80 * YS;        // 80*AS
  float* tBuf = attP + 80 * AS;      // 8 * 80*TS

  const int b = blockIdx.x;
  const int tid = threadIdx.x;
  const int wave = tid >> 5;         // wave32
  const int lane = tid & 31;
  const int l16 = lane & 15;
  const int hi = lane >> 4;
  const int koff = hi * 2;           // K offset for A/B operands (16x16x4)
  const int rbase = hi * 8;          // C/D row offset

  const float* xb = xin + (size_t)b * 66 * K;
  for (int i = tid; i < 80 * K; i += 256) {
    int r = i / K, c = i - r * K;
    yP[r * YS + c] = (r < 66) ? xb[r * K + c] : 0.0f;
  }
  for (int i = tid; i < 80 * 80; i += 256) {
    int r = i / 80, c = i - r * 80;
    attP[r * AS + c] = (r < 66 && c < 66) ? att[r * 66 + c] : 0.0f;
  }
  __syncthreads();

  float* tw = tBuf + wave * (80 * TS);
  constexpr int nSlabs = (N + 15) >> 4;

  for (int slab = wave; slab < nSlabs; slab += 8) {
    const int n0 = slab << 4;
    int colW = n0 + l16;
    if (colW >= N) colW = N - 1;     // clamp; discarded in epilogue
    const float* wp = W + koff * N + colW;
    const float* ap = yP + l16 * YS + koff;

    // ---- GEMM1: t(80x16) = yP(80xK) @ W(:,slab) --------------------------
    v8f acc0 = {}, acc1 = {}, acc2 = {}, acc3 = {}, acc4 = {};
#pragma unroll 4
    for (int k = 0; k < K; k += 4) {
      v2f bOp;
      bOp.x = wp[k * N];
      bOp.y = wp[(k + 1) * N];
      v2f a0 = *(const v2f*)&ap[0 * 16 * YS + k];
      v2f a1 = *(const v2f*)&ap[1 * 16 * YS + k];
      v2f a2 = *(const v2f*)&ap[2 * 16 * YS + k];
      v2f a3 = *(const v2f*)&ap[3 * 16 * YS + k];
      v2f a4 = *(const v2f*)&ap[4 * 16 * YS + k];
      acc0 = wmma_f32(a0, bOp, acc0);
      acc1 = wmma_f32(a1, bOp, acc1);
      acc2 = wmma_f32(a2, bOp, acc2);
      acc3 = wmma_f32(a3, bOp, acc3);
      acc4 = wmma_f32(a4, bOp, acc4);
    }
    // spill t to wave-private LDS (padded 80x20 region)
    {
      v8f accs[5] = {acc0, acc1, acc2, acc3, acc4};
#pragma unroll
      for (int m = 0; m < 5; ++m)
#pragma unroll
        for (int g = 0; g < 8; ++g)
          tw[(m * 16 + rbase + g) * TS + l16] = accs[m][g];
    }
    asm volatile("s_wait_dscnt 0x0" ::: "memory");  // wave-internal RAW

    // ---- GEMM2: o(80x16) = attP(80x80) @ t(80x16) ------------------------
    v8f o0 = {}, o1 = {}, o2 = {}, o3 = {}, o4 = {};
    const float* bp = tw + koff * TS + l16;
    const float* cp = attP + l16 * AS + koff;
#pragma unroll 4
    for (int k = 0; k < 80; k += 4) {
      v2f bOp;
      bOp.x = bp[k * TS];
      bOp.y = bp[(k + 1) * TS];
      v2f a0 = *(const v2f*)&cp[0 * 16 * AS + k];
      v2f a1 = *(const v2f*)&cp[1 * 16 * AS + k];
      v2f a2 = *(const v2f*)&cp[2 * 16 * AS + k];
      v2f a3 = *(const v2f*)&cp[3 * 16 * AS + k];
      v2f a4 = *(const v2f*)&cp[4 * 16 * AS + k];
      o0 = wmma_f32(a0, bOp, o0);
      o1 = wmma_f32(a1, bOp, o1);
      o2 = wmma_f32(a2, bOp, o2);
      o3 = wmma_f32(a3, bOp, o3);
      o4 = wmma_f32(a4, bOp, o4);
    }

    // ---- epilogue --------------------------------------------------------
    v8f outs[5] = {o0, o1, o2, o3, o4};
    const int col = n0 + l16;
#pragma unroll
    for (int m = 0; m < 5; ++m) {
#pragma unroll
      for (int g = 0; g < 8; ++g) {
        int row = m * 16 + rbase + g;
        if (row < 66 && col < N) {
          float v = outs[m][g] + bias[col];
          if (BN) {
            int gi = row * N + col;
            v = tanhf(gamma[gi] * v * BN_INV_F + beta[gi]);
          }
          size_t oi = ((size_t)b * 66 + row) * N + col;
          if (RES) v += resid[oi];
          out[oi] = v;
        }
      }
    }
  }
}

// ---------------------------------------------------------------------------
// Stage 5: out[b,v,0,f] = sum_{d<10} dctm[d,v]*dct_out[b,f,d]
// ---------------------------------------------------------------------------
__global__ void idct_kernel(const float* __restrict__ dct_out,  // (B,66,20)
                            const float* __restrict__ dctm,     // (35,35)
                            float* __restrict__ out,            // (B,35,66)
                            int total)
{
  int idx = blockIdx.x * blockDim.x + threadIdx.x;
  if (idx >= total) return;
  int f = idx % 66;
  int v = (idx / 66) % 35;
  int b = idx / (35 * 66);
  const float* dd = dct_out + ((size_t)b * 66 + f) * 20;
  float a = 0.f;
#pragma unroll
  for (int d = 0; d < 10; ++d) a += dctm[d * 35 + v] * dd[d];
  out[idx] = a;
}

// ---------------------------------------------------------------------------
static constexpr size_t lds_bytes(int K) {
  return (size_t)(80 * (K + 4) + 80 * 84 + 8 * 80 * 20) * 4;
}

extern "C" void kernel_launch(void* const* d_in, const int* in_sizes, int n_in,
                              void* d_out, int out_size, void* d_ws,
                              size_t ws_size, hipStream_t stream) {
  const float* src    = (const float*)d_in[0];   // (B,50,66)
  const float* q_w    = (const float*)d_in[1];
  const float* q_b    = (const float*)d_in[2];
  const float* k_w    = (const float*)d_in[3];
  const float* k_b    = (const float*)d_in[4];
  const float* att_in = (const float*)d_in[5];
  const float* w_in   = (const float*)d_in[6];   // (20,512)
  const float* b_in   = (const float*)d_in[7];
  const float* g_in   = (const float*)d_in[8];
  const float* be_in  = (const float*)d_in[9];
  const float* A1     = (const float*)d_in[10];  // (2,66,66)
  const float* W1     = (const float*)d_in[11];  // (2,512,512)
  const float* B1     = (const float*)d_in[12];  // (2,512)
  const float* G1     = (const float*)d_in[13];  // (2,33792)
  const float* Be1    = (const float*)d_in[14];
  const float* A2     = (const float*)d_in[15];
  const float* W2     = (const float*)d_in[16];
  const float* B2     = (const float*)d_in[17];
  const float* G2     = (const float*)d_in[18];
  const float* Be2    = (const float*)d_in[19];
  const float* attO   = (const float*)d_in[20];
  const float* w_out  = (const float*)d_in[21];  // (512,20)
  const float* b_out  = (const float*)d_in[22];
  float* out = (float*)d_out;

  const int B = in_sizes[0] / (50 * 66);  // 2048
  const int D = 512, F = 66;

  // workspace layout (floats)
  float* ws = (float*)d_ws;
  float* dctm    = ws;                 size_t off = 1280;
  float* dct_in  = ws + off;           off += (size_t)B * F * 20;
  float* dct_out = ws + off;           off += (size_t)B * F * 20;
  float* yA      = ws + off;           off += (size_t)B * F * D;
  float* yB      = ws + off;

  const size_t lds512 = lds_bytes(512);  // 243,200 B (< 320KB WGP LDS)
  const size_t lds20  = lds_bytes(20);
  (void)hipFuncSetAttribute((const void*)gconv_kernel<20, 512, true, false>,
                            hipFuncAttributeMaxDynamicSharedMemorySize,
                            (int)lds20);
  (void)hipFuncSetAttribute((const void*)gconv_kernel<512, 512, true, false>,
                            hipFuncAttributeMaxDynamicSharedMemorySize,
                            (int)lds512);
  (void)hipFuncSetAttribute((const void*)gconv_kernel<512, 512, true, true>,
                            hipFuncAttributeMaxDynamicSharedMemorySize,
                            (int)lds512);
  (void)hipFuncSetAttribute((const void*)gconv_kernel<512, 20, false, true>,
                            hipFuncAttributeMaxDynamicSharedMemorySize,
                            (int)lds512);

  init_dctm_kernel<<<(35 * 35 + 255) / 256, 256, 0, stream>>>(dctm);
  stage1_kernel<<<B, 256, 0, stream>>>(src, q_w, q_b, k_w, k_b, dctm, dct_in);

  // input gconv: dct_in(66x20) -> yA(66x512), BN+tanh
  gconv_kernel<20, 512, true, false><<<B, 256, lds20, stream>>>(
      dct_in, att_in, w_in, b_in, g_in, be_in, nullptr, yA);
  // 2 residual blocks; residual add folded into second gconv (out aliases
  // resid element-wise: each element is read then written by one lane).
  for (int i = 0; i < 2; ++i) {
    gconv_kernel<512, 512, true, false><<<B, 256, lds512, stream>>>(
        yA, A1 + (size_t)i * F * F, W1 + (size_t)i * D * D, B1 + (size_t)i * D,
        G1 + (size_t)i * F * D, Be1 + (size_t)i * F * D, nullptr, yB);
    gconv_kernel<512, 512, true, true><<<B, 256, lds512, stream>>>(
        yB, A2 + (size_t)i * F * F, W2 + (size_t)i * D * D, B2 + (size_t)i * D,
        G2 + (size_t)i * F * D, Be2 + (size_t)i * F * D, yA, yA);
  }
  // output gconv: yA(66x512) -> dct_out(66x20), + dct_in residual, no BN
  gconv_kernel<512, 20, false, true><<<B, 256, lds512, stream>>>(
      yA, attO, w_out, b_out, nullptr, nullptr, dct_in, dct_out);
  // IDCT to (B,35,1,66)
  int total = B * 35 * 66;
  idct_kernel<<<(total + 255) / 256, 256, 0, stream>>>(dct_out, dctm, out,
                                                       total);
}